// CannyFilter_3891240370193
// MI455X (gfx1250) — compile-verified
//
#include <hip/hip_runtime.h>
#include <stdint.h>

// ---------------------------------------------------------------------------
// Fused Canny for MI455X (gfx1250, wave32).
// Memory-bound stencil (~150MB traffic => ~6.5us @ 23.3 TB/s). Data movement
// is the whole game: B128 async DMA (GLOBAL_LOAD_ASYNC_TO_LDS_B128) stages the
// halo'd RGB tile into LDS, the 7-stage pipeline stays LDS-resident, and the
// 4 output planes are written as float4 (global_store_b128).
// ---------------------------------------------------------------------------

#define TILE   32
#define HALOY  5
#define HALOX  8            // x halo padded to 8 so 4-elem groups stay aligned
#define LH     42           // TILE + 2*HALOY rows
#define LWX    48           // TILE + 2*HALOX cols == LDS pitch (floats)
#define NGRP   12           // LWX / 4 b128 groups per row
#define NTHREADS 256        // 8 wave32 per workgroup

__device__ __forceinline__ uint32_t lds_byte_off(const void* p) {
  // generic ptr to __shared__ : low 32 bits are the workgroup LDS byte offset
  return (uint32_t)(uintptr_t)p;
}

__device__ __forceinline__ void async_ld_b128(uint32_t lds_off, const float* g) {
  asm volatile("global_load_async_to_lds_b128 %0, %1, off"
               :: "v"(lds_off), "v"((unsigned long long)(uintptr_t)g)
               : "memory");
}

__device__ __forceinline__ int clampi(int v, int lo, int hi) {
  return v < lo ? lo : (v > hi ? hi : v);
}

__global__ __launch_bounds__(NTHREADS)
void canny_fused_kernel(const float* __restrict__ img,
                        float* __restrict__ out0, float* __restrict__ out1,
                        float* __restrict__ out2, float* __restrict__ out3) {
  __shared__ __align__(16) float sA[LH * LWX];  // R stage -> blur -> thin2
  __shared__ __align__(16) float sB[LH * LWX];  // G stage -> grad_x
  __shared__ __align__(16) float sC[LH * LWX];  // B stage -> grad_y
  __shared__ __align__(16) float sD[LH * LWX];  // gray    -> grad magnitude
  __shared__ __align__(16) float sE[LH * LWX];  // thin1 ({0, .5, 1})

  const int tid = threadIdx.x;
  const int b   = blockIdx.z;
  const int ox  = blockIdx.x * TILE;
  const int oy  = blockIdx.y * TILE;

  const size_t plane = 512 * 512;
  const float* imR = img + (size_t)b * 3 * plane;
  const float* imG = imR + plane;
  const float* imB = imG + plane;

  const uint32_t bA = lds_byte_off(sA);
  const uint32_t bB = lds_byte_off(sB);
  const uint32_t bC = lds_byte_off(sC);

  // ---- Phase 0: async B128 DMA of the 42x48 halo'd R,G,B tile into LDS ----
  // x-groups are 4-aligned with the image => each group is fully in-image or
  // fully OOB; OOB groups get a clamped (safe) address and are masked later.
  for (int i = tid; i < LH * NGRP; i += NTHREADS) {
    int ly = i / NGRP, grp = i - ly * NGRP;
    int gy  = oy - HALOY + ly;
    int gx0 = ox - HALOX + grp * 4;
    int cy  = clampi(gy, 0, 511);
    int cx0 = clampi(gx0, 0, 508);
    size_t o = (size_t)cy * 512 + (size_t)cx0;
    uint32_t l = 16u * (uint32_t)(ly * NGRP + grp);
    async_ld_b128(bA + l, imR + o);
    async_ld_b128(bB + l, imG + o);
    async_ld_b128(bC + l, imB + o);
  }
  asm volatile("s_wait_asynccnt 0" ::: "memory");
  __syncthreads();

  // ---- Phase 1: grayscale on [0,42)x[3,45); zero outside image ----
  for (int i = tid; i < 42 * 42; i += NTHREADS) {
    int ly = i / 42, lx = i - ly * 42 + 3;
    int gy = oy - HALOY + ly, gx = ox - HALOX + lx;
    bool in = ((unsigned)gy < 512u) & ((unsigned)gx < 512u);
    int c = ly * LWX + lx;
    float g = 0.299f * sA[c] + 0.587f * sB[c] + 0.114f * sC[c];
    sD[c] = in ? g : 0.0f;
  }
  __syncthreads();

  // ---- Phase 2: 3x3 gaussian -> sA on [1,41)x[4,44); zero outside image ----
  const float WC = 0.20418000f, WE = 0.12384141f, WK = 0.07511361f;
  for (int i = tid; i < 40 * 40; i += NTHREADS) {
    int ly = i / 40, lx = i - ly * 40;
    ly += 1; lx += 4;
    int gy = oy - HALOY + ly, gx = ox - HALOX + lx;
    bool in = ((unsigned)gy < 512u) & ((unsigned)gx < 512u);
    int c = ly * LWX + lx;
    float v = WC * sD[c]
            + WE * (sD[c - 1] + sD[c + 1] + sD[c - LWX] + sD[c + LWX])
            + WK * (sD[c - LWX - 1] + sD[c - LWX + 1] + sD[c + LWX - 1] + sD[c + LWX + 1]);
    sA[c] = in ? v : 0.0f;
  }
  __syncthreads();

  // ---- Phase 3: sobel + magnitude on [2,40)x[5,43): sB=gx, sC=gy, sD=mag ----
  for (int i = tid; i < 38 * 38; i += NTHREADS) {
    int ly = i / 38, lx = i - ly * 38;
    ly += 2; lx += 5;
    int c = ly * LWX + lx;
    float l0 = sA[c - LWX - 1], u  = sA[c - LWX], r0 = sA[c - LWX + 1];
    float l1 = sA[c - 1],                          r1 = sA[c + 1];
    float l2 = sA[c + LWX - 1], dn = sA[c + LWX], r2 = sA[c + LWX + 1];
    float gxv = 0.5f * (r0 - l0) + (r1 - l1) + 0.5f * (r2 - l2);
    float gyv = 0.5f * (l2 - l0) + (dn - u)  + 0.5f * (r2 - r0);
    sB[c] = gxv;
    sC[c] = gyv;
    sD[c] = sqrtf(gxv * gxv + gyv * gyv);
  }
  __syncthreads();

  // ---- Phase 4: NMS (edge-replicate) + double threshold -> sE on [3,39)x[6,42) ----
  // d = round(atan2(gy,gx)/45deg) mod 4 via octant compares (no atan2f):
  {
    const int dy_t[4] = {0, -1, -1, -1};
    const int dx_t[4] = {1,  1,  0, -1};
    for (int i = tid; i < 36 * 36; i += NTHREADS) {
      int ly = i / 36, lx = i - ly * 36;
      ly += 3; lx += 6;
      int gy = oy - HALOY + ly, gx = ox - HALOX + lx;
      bool in = ((unsigned)gy < 512u) & ((unsigned)gx < 512u);
      int c = ly * LWX + lx;
      float t1 = 0.0f;
      if (in) {
        float gxv = sB[c], gyv = sC[c], m = sD[c];
        float ax = fabsf(gxv), ay = fabsf(gyv);
        int d;
        if (ay <= 0.41421356f * ax)      d = 0;
        else if (ay >= 2.41421356f * ax) d = 2;
        else d = ((gxv > 0.0f) == (gyv > 0.0f)) ? 1 : 3;
        int dy = dy_t[d], dx = dx_t[d];
        int py = clampi(gy + dy, 0, 511), px = clampi(gx + dx, 0, 511);
        int qy = clampi(gy - dy, 0, 511), qx = clampi(gx - dx, 0, 511);
        float mp = sD[(py - oy + HALOY) * LWX + (px - ox + HALOX)];
        float mn = sD[(qy - oy + HALOY) * LWX + (qx - ox + HALOX)];
        float t0 = (fminf(m - mp, m - mn) > 0.0f) ? m : 0.0f;
        t1 = (t0 > 0.1f ? 0.5f : 0.0f) + (t0 > 0.3f ? 0.5f : 0.0f);
      }
      sE[c] = t1;
    }
  }
  __syncthreads();

  // ---- Phase 5: hysteresis -> sA on [4,38)x[7,41) ----
  for (int i = tid; i < 34 * 34; i += NTHREADS) {
    int ly = i / 34, lx = i - ly * 34;
    ly += 4; lx += 7;
    int gy = oy - HALOY + ly, gx = ox - HALOX + lx;
    bool in = ((unsigned)gy < 512u) & ((unsigned)gx < 512u);
    int c = ly * LWX + lx;
    float t1 = sE[c];
    float mx = sE[c - LWX - 1];
    mx = fmaxf(mx, sE[c - LWX]); mx = fmaxf(mx, sE[c - LWX + 1]);
    mx = fmaxf(mx, sE[c - 1]);   mx = fmaxf(mx, sE[c]);
    mx = fmaxf(mx, sE[c + 1]);
    mx = fmaxf(mx, sE[c + LWX - 1]); mx = fmaxf(mx, sE[c + LWX]);
    mx = fmaxf(mx, sE[c + LWX + 1]);
    float t2 = (t1 == 1.0f) ? 1.0f : ((t1 == 0.5f && mx == 1.0f) ? 1.0f : 0.0f);
    sA[c] = in ? t2 : 0.0f;
  }
  __syncthreads();

  // ---- Phase 6: final 3x3 dilation + float4 stores to all 4 planes ----
  // 256 threads x 4 consecutive x-pixels == 32x32 tile, one pass.
  {
    int ly  = (tid >> 3) + HALOY;          // 5..36
    int lx0 = (tid & 7) * 4 + HALOX;       // 8,12,...,36
    int c   = ly * LWX + lx0;
    // column maxes over 3 rows for the 6 columns covering 4 windows
    float cm[6];
#pragma unroll
    for (int j = 0; j < 6; ++j) {
      int cc = c - 1 + j;
      float v = fmaxf(sA[cc - LWX], sA[cc]);
      cm[j] = fmaxf(v, sA[cc + LWX]);
    }
    float4 t4, m4, x4, y4;
    float* tp = &t4.x; float* mp = &m4.x; float* xp = &x4.x; float* yp = &y4.x;
#pragma unroll
    for (int j = 0; j < 4; ++j) {
      float t = fmaxf(fmaxf(cm[j], cm[j + 1]), cm[j + 2]);
      tp[j] = t;
      mp[j] = sD[c + j] * t;
      xp[j] = sB[c + j] * t;
      yp[j] = sC[c + j] * t;
    }
    int gy = oy + (ly - HALOY), gx0 = ox + (lx0 - HALOX);
    size_t o = ((size_t)b << 18) + ((size_t)gy << 9) + (size_t)gx0;
    *reinterpret_cast<float4*>(out0 + o) = t4;
    *reinterpret_cast<float4*>(out1 + o) = m4;
    *reinterpret_cast<float4*>(out2 + o) = x4;
    *reinterpret_cast<float4*>(out3 + o) = y4;
  }
}

extern "C" void kernel_launch(void* const* d_in, const int* in_sizes, int n_in,
                              void* d_out, int out_size, void* d_ws, size_t ws_size,
                              hipStream_t stream) {
  (void)in_sizes; (void)n_in; (void)out_size; (void)d_ws; (void)ws_size;
  const float* img = (const float*)d_in[0];
  float* o = (float*)d_out;
  const size_t n = (size_t)16 * 512 * 512;   // elements per output plane
  dim3 grid(512 / TILE, 512 / TILE, 16);
  dim3 block(NTHREADS);
  canny_fused_kernel<<<grid, block, 0, stream>>>(img, o, o + n, o + 2 * n, o + 3 * n);
}